// PteCriterion_2336462209674
// MI455X (gfx1250) — compile-verified
//
#include <hip/hip_runtime.h>
#include <math.h>

typedef __attribute__((ext_vector_type(2))) float v2f;
typedef __attribute__((ext_vector_type(8))) float v8f;

#define B_ 32
#define S_ 128
#define V_ 50257
#define C_ 16
#define K_ 128   // C_*F_ flattened reduction dim
#define LP_ 132  // padded LDS row stride (132 % 64 = 4 -> conflict-free strided reads)

__launch_bounds__(128)
__global__ void pte_criterion_kernel(const float* __restrict__ logits,
                                     const int* __restrict__ mlm_labels,
                                     const int* __restrict__ labels,
                                     const float* __restrict__ weight,
                                     const int* __restrict__ m2c,
                                     const float* __restrict__ filler_len,
                                     float* __restrict__ out) {
  __shared__ float G[B_][LP_];     // gathered logits, A matrix (32 x 128)
  __shared__ float BdT[C_][LP_];   // transposed block-diag weights: BdT[c][k]
  __shared__ float CL[B_][C_];     // cls_logits
  __shared__ int   posArr[B_];
  __shared__ float lp[B_];         // per-sample log-prob at label
  __shared__ int   pred[B_];

  const int t = threadIdx.x;

  // ---- Stage 1: pos[b] = first s with mlm_labels[b,s] >= 0 (argmax of bool) ----
  if (t < B_) {
    int p = 0;
    for (int s = 0; s < S_; ++s) {
      if (mlm_labels[t * S_ + s] >= 0) { p = s; break; }
    }
    posArr[t] = p;
  }
  __syncthreads();

  // ---- Stage 2a: gather A: G[b][k] = logits[b, pos[b], max(m2c[k],0)] ----
  for (int i = t; i < B_ * K_; i += 128) {
    int b = i >> 7;
    int k = i & (K_ - 1);
    int idx = m2c[k];
    if (idx < 0) idx = 0;
    const float* row = logits + ((size_t)b * S_ + (size_t)posArr[b]) * (size_t)V_;
    G[b][k] = row[idx];
  }

  // ---- Stage 2b: build B^T: BdT[c][k] = ((k>>3)==c && m2c[k]>0) ? weight[k] : 0 ----
  for (int i = t; i < C_ * K_; i += 128) {
    int c = i >> 7;
    int k = i & (K_ - 1);
    float w = ((k >> 3) == c && m2c[k] > 0) ? weight[k] : 0.0f;
    BdT[c][k] = w;
  }
  __syncthreads();

  // ---- Stage 3: cls = G(32x128) x Bd(128x16) via V_WMMA_F32_16X16X4_F32 ----
  // Wave 0 -> output rows 0..15, wave 1 -> rows 16..31. EXEC all-1s per wave.
  if (t < 64) {
    const int lane     = t & 31;
    const int tileBase = (t >> 5) << 4;
    const int n        = lane & 15;            // output column = class c
    const int khi      = (lane >= 16) ? 2 : 0; // K sub-slot per half-wave
    const int arow     = tileBase + (lane & 15);

    const v2f* __restrict__ aRow = (const v2f*)&G[arow][0];   // k0 even -> 8B aligned
    const v2f* __restrict__ bRow = (const v2f*)&BdT[n][0];

    v8f acc = {0.f, 0.f, 0.f, 0.f, 0.f, 0.f, 0.f, 0.f};
    #pragma unroll 4
    for (int kk = 0; kk < K_ / 4; ++kk) {
      const int p = kk * 2 + (khi >> 1);       // v2f index of {k0, k0+1}
      v2f a  = aRow[p];
      v2f bf = bRow[p];
      acc = __builtin_amdgcn_wmma_f32_16x16x4_f32(false, a, false, bf,
                                                  (short)0, acc, false, false);
    }
    const float fl = filler_len[n];
    for (int r = 0; r < 8; ++r) {
      const int m = tileBase + r + ((lane >= 16) ? 8 : 0);
      CL[m][n] = acc[r] / fl;   // divide-after-sum, matching the reference
    }
  }
  __syncthreads();

  // ---- Stage 4: per-sample argmax (first-max tiebreak) + log-softmax at label ----
  if (t < B_) {
    float mx = CL[t][0];
    int am = 0;
    for (int c = 1; c < C_; ++c) {
      float v = CL[t][c];
      if (v > mx) { mx = v; am = c; }
    }
    float se = 0.f;
    for (int c = 0; c < C_; ++c) se += expf(CL[t][c] - mx);
    const int lab = labels[t];
    lp[t]   = CL[t][lab] - mx - logf(se);
    pred[t] = am;
  }
  __syncthreads();

  // ---- Stage 5: reduce loss, emit outputs (loss, then 32 predictions) ----
  if (t == 0) {
    float s = 0.f;
    for (int b = 0; b < B_; ++b) s += lp[b];
    out[0] = -s / (float)B_;
  }
  if (t < B_) out[1 + t] = (float)pred[t];
}

extern "C" void kernel_launch(void* const* d_in, const int* in_sizes, int n_in,
                              void* d_out, int out_size, void* d_ws, size_t ws_size,
                              hipStream_t stream) {
  const float* logits     = (const float*)d_in[0];
  const int*   mlm_labels = (const int*)d_in[1];
  const int*   labels     = (const int*)d_in[2];
  const float* weight     = (const float*)d_in[3];
  const int*   m2c        = (const int*)d_in[4];
  const float* filler_len = (const float*)d_in[5];
  float*       out        = (float*)d_out;

  hipLaunchKernelGGL(pte_criterion_kernel, dim3(1), dim3(128), 0, stream,
                     logits, mlm_labels, labels, weight, m2c, filler_len, out);
}